// GCN_model_17008070492799
// MI455X (gfx1250) — compile-verified
//
#include <hip/hip_runtime.h>

typedef __attribute__((ext_vector_type(2))) float v2f;
typedef __attribute__((ext_vector_type(8))) float v8f;

#define NFEAT 128
#define HID   16
#define NGRAPH 64

// ---------------- degree / norm precompute ----------------

__global__ void k_deg_init(float* __restrict__ deg, int n) {
  int i = blockIdx.x * blockDim.x + threadIdx.x;
  if (i < n) deg[i] = 1.0f;               // self-loop contributes 1 to every node
}

__global__ void k_deg_count(const int* __restrict__ dst, float* __restrict__ deg, int ne) {
  int e = blockIdx.x * blockDim.x + threadIdx.x;
  if (e < ne) atomicAdd(&deg[dst[e]], 1.0f);
}

__global__ void k_dinv(float* __restrict__ deg, int n) {
  int i = blockIdx.x * blockDim.x + threadIdx.x;
  if (i < n) {
    float d = deg[i];
    deg[i] = (d > 0.0f) ? rsqrtf(d) : 0.0f;   // in-place deg -> deg^-1/2
  }
}

__global__ void k_norm(const int* __restrict__ src, const int* __restrict__ dst,
                       const float* __restrict__ dinv, float* __restrict__ norm, int ne) {
  int e = blockIdx.x * blockDim.x + threadIdx.x;
  if (e < ne) norm[e] = dinv[src[e]] * dinv[dst[e]];
}

// ---------------- fp32 WMMA GEMM: H[n,NOUT] = prelu(A)[n,K] @ W[K,NOUT] ----------------
// One wave computes one 16-row tile via V_WMMA_F32_16X16X4_F32 (K stepped by 4).
// NOUT<16 handled by zero-padding B columns in registers.

template<int K, int NOUT, bool PRELU>
__global__ void k_gemm_wmma(const float* __restrict__ A, const float* __restrict__ W,
                            const float* __restrict__ alpha, float* __restrict__ H,
                            int n_nodes) {
  const int lane = threadIdx.x & 31;
  const int wave = threadIdx.x >> 5;
  const int tile = blockIdx.x * (blockDim.x >> 5) + wave;
  if (tile * 16 >= n_nodes) return;       // wave-uniform: EXEC stays all-ones for WMMA
  const int half = lane >> 4;             // 0: K={k,k+1}, 1: K={k+2,k+3}
  const int idx  = lane & 15;             // M (for A) / N (for B,D)
  const float sl = PRELU ? alpha[0] : 0.0f;

  const float* arow = A + (size_t)(tile * 16 + idx) * K + 2 * half;
  v8f c = {};
#pragma unroll
  for (int k = 0; k < K; k += 4) {
    float ax = arow[k + 0];
    float ay = arow[k + 1];
    if (PRELU) {
      ax = ax > 0.0f ? ax : sl * ax;
      ay = ay > 0.0f ? ay : sl * ay;
    }
    v2f a; a.x = ax; a.y = ay;
    v2f b;
    b.x = (idx < NOUT) ? W[(k + 2 * half + 0) * NOUT + idx] : 0.0f;
    b.y = (idx < NOUT) ? W[(k + 2 * half + 1) * NOUT + idx] : 0.0f;
    c = __builtin_amdgcn_wmma_f32_16x16x4_f32(false, a, false, b, (short)0, c, false, false);
  }
  // D layout: VGPR j -> row (j + 8*half), col idx
#pragma unroll
  for (int j = 0; j < 8; ++j) {
    int m = tile * 16 + j + 8 * half;
    if (idx < NOUT) H[(size_t)m * NOUT + idx] = c[j];
  }
}

// ---------------- aggregation: out = bias + dinv^2 * h  (self loop), then edge scatter ----------------

__global__ void k_agg_init16(const float* __restrict__ H, const float* __restrict__ bias,
                             const float* __restrict__ dinv, float* __restrict__ Out, int n) {
  int t = blockIdx.x * blockDim.x + threadIdx.x;
  int i = t >> 2, q = t & 3;
  if (i >= n) return;
  float di = dinv[i], s = di * di;
  float4 h = *(const float4*)(H + (size_t)i * 16 + q * 4);
  float4 b = *(const float4*)(bias + q * 4);
  float4 o;
  o.x = b.x + s * h.x; o.y = b.y + s * h.y; o.z = b.z + s * h.z; o.w = b.w + s * h.w;
  *(float4*)(Out + (size_t)i * 16 + q * 4) = o;
}

__global__ void k_agg_edges16(const int* __restrict__ src, const int* __restrict__ dst,
                              const float* __restrict__ norm, const float* __restrict__ H,
                              float* __restrict__ Out, int ne) {
  int t = blockIdx.x * blockDim.x + threadIdx.x;
  int e = t >> 2, q = t & 3;
  if (e >= ne) return;
  int s = src[e], d = dst[e];
  float w = norm[e];
  float4 h = *(const float4*)(H + (size_t)s * 16 + q * 4);
  float* o = Out + (size_t)d * 16 + q * 4;
  atomicAdd(o + 0, h.x * w);
  atomicAdd(o + 1, h.y * w);
  atomicAdd(o + 2, h.z * w);
  atomicAdd(o + 3, h.w * w);
}

__global__ void k_agg_init2(const float* __restrict__ H, const float* __restrict__ bias,
                            const float* __restrict__ dinv, float* __restrict__ Out, int n) {
  int i = blockIdx.x * blockDim.x + threadIdx.x;
  if (i >= n) return;
  float di = dinv[i], s = di * di;
  float2 h = *(const float2*)(H + (size_t)i * 2);
  float2 o;
  o.x = bias[0] + s * h.x; o.y = bias[1] + s * h.y;
  *(float2*)(Out + (size_t)i * 2) = o;
}

__global__ void k_agg_edges2(const int* __restrict__ src, const int* __restrict__ dst,
                             const float* __restrict__ norm, const float* __restrict__ H,
                             float* __restrict__ Out, int ne) {
  int e = blockIdx.x * blockDim.x + threadIdx.x;
  if (e >= ne) return;
  int s = src[e], d = dst[e];
  float w = norm[e];
  float2 h = *(const float2*)(H + (size_t)s * 2);
  float* o = Out + (size_t)d * 2;
  atomicAdd(o + 0, h.x * w);
  atomicAdd(o + 1, h.y * w);
}

// ---------------- global mean pool ----------------

__global__ void k_pool_init(float* __restrict__ sums, float* __restrict__ cnts) {
  int i = threadIdx.x;
  if (i < NGRAPH * 2) sums[i] = 0.0f;
  if (i < NGRAPH) cnts[i] = 0.0f;
}

__global__ void k_pool_acc(const float* __restrict__ H, const int* __restrict__ batch,
                           float* __restrict__ sums, float* __restrict__ cnts, int n) {
  int i = blockIdx.x * blockDim.x + threadIdx.x;
  if (i >= n) return;
  int g = batch[i];
  atomicAdd(&sums[g * 2 + 0], H[(size_t)i * 2 + 0]);
  atomicAdd(&sums[g * 2 + 1], H[(size_t)i * 2 + 1]);
  atomicAdd(&cnts[g], 1.0f);
}

__global__ void k_pool_final(const float* __restrict__ sums, const float* __restrict__ cnts,
                             float* __restrict__ out) {
  int i = threadIdx.x;
  if (i < NGRAPH * 2) {
    float c = cnts[i >> 1];
    out[i] = sums[i] / fmaxf(c, 1.0f);
  }
}

// ---------------- launcher ----------------

static inline size_t align256(size_t x) { return (x + 255) & ~(size_t)255; }

extern "C" void kernel_launch(void* const* d_in, const int* in_sizes, int n_in,
                              void* d_out, int out_size, void* d_ws, size_t ws_size,
                              hipStream_t stream) {
  const float* x     = (const float*)d_in[0];
  const int*   eidx  = (const int*)d_in[1];
  const int*   batch = (const int*)d_in[2];
  const float* W0 = (const float*)d_in[3];  const float* b0 = (const float*)d_in[4];
  const float* a0 = (const float*)d_in[5];
  const float* W1 = (const float*)d_in[6];  const float* b1 = (const float*)d_in[7];
  const float* a1 = (const float*)d_in[8];
  const float* W2 = (const float*)d_in[9];  const float* b2 = (const float*)d_in[10];
  const float* a2 = (const float*)d_in[11];
  const float* W3 = (const float*)d_in[12]; const float* b3 = (const float*)d_in[13];
  float* out = (float*)d_out;

  const int n  = in_sizes[0] / NFEAT;   // 100000
  const int ne = in_sizes[1] / 2;       // 3200000
  const int* src = eidx;
  const int* dst = eidx + ne;

  char* ws = (char*)d_ws;
  size_t off = 0;
  float* dinv = (float*)(ws + off); off += align256((size_t)n * 4);
  float* norm = (float*)(ws + off); off += align256((size_t)ne * 4);
  float* buf0 = (float*)(ws + off); off += align256((size_t)n * HID * 4);
  float* buf1 = (float*)(ws + off); off += align256((size_t)n * HID * 4);
  float* h3   = (float*)(ws + off); off += align256((size_t)n * 2 * 4);
  float* a3   = (float*)(ws + off); off += align256((size_t)n * 2 * 4);
  float* psum = (float*)(ws + off); off += align256((size_t)NGRAPH * 2 * 4);
  float* pcnt = (float*)(ws + off);

  const int B = 256;
  const int gN   = (n + B - 1) / B;
  const int gE   = (ne + B - 1) / B;
  const int gE4  = (int)(((size_t)ne * 4 + B - 1) / B);
  const int gN4  = (int)(((size_t)n * 4 + B - 1) / B);
  const int tiles = n / 16;                       // 6250, exact
  const int gT   = (tiles + 3) / 4;               // 4 waves per 128-thread block

  // degrees -> dinv -> per-edge norm
  k_deg_init<<<gN, B, 0, stream>>>(dinv, n);
  k_deg_count<<<gE, B, 0, stream>>>(dst, dinv, ne);
  k_dinv<<<gN, B, 0, stream>>>(dinv, n);
  k_norm<<<gE, B, 0, stream>>>(src, dst, dinv, norm, ne);

  // layer 0: H0 = X @ W0 ; A0 = b0 + scatter(norm * H0[src])
  k_gemm_wmma<NFEAT, HID, false><<<gT, 128, 0, stream>>>(x, W0, nullptr, buf0, n);
  k_agg_init16<<<gN4, B, 0, stream>>>(buf0, b0, dinv, buf1, n);
  k_agg_edges16<<<gE4, B, 0, stream>>>(src, dst, norm, buf0, buf1, ne);

  // layer 1: H1 = prelu(A0,a0) @ W1 ; aggregate
  k_gemm_wmma<HID, HID, true><<<gT, 128, 0, stream>>>(buf1, W1, a0, buf0, n);
  k_agg_init16<<<gN4, B, 0, stream>>>(buf0, b1, dinv, buf1, n);
  k_agg_edges16<<<gE4, B, 0, stream>>>(src, dst, norm, buf0, buf1, ne);

  // layer 2
  k_gemm_wmma<HID, HID, true><<<gT, 128, 0, stream>>>(buf1, W2, a1, buf0, n);
  k_agg_init16<<<gN4, B, 0, stream>>>(buf0, b2, dinv, buf1, n);
  k_agg_edges16<<<gE4, B, 0, stream>>>(src, dst, norm, buf0, buf1, ne);

  // layer 3: H3 = prelu(A2,a2) @ W3 (16->2) ; aggregate (no prelu after)
  k_gemm_wmma<HID, 2, true><<<gT, 128, 0, stream>>>(buf1, W3, a2, h3, n);
  k_agg_init2<<<gN, B, 0, stream>>>(h3, b3, dinv, a3, n);
  k_agg_edges2<<<gE, B, 0, stream>>>(src, dst, norm, h3, a3, ne);

  // global mean pool
  k_pool_init<<<1, 128, 0, stream>>>(psum, pcnt);
  k_pool_acc<<<gN, B, 0, stream>>>(a3, batch, psum, pcnt, n);
  k_pool_final<<<1, 128, 0, stream>>>(psum, pcnt, out);
}